// NodeClassifier_54425825575074
// MI455X (gfx1250) — compile-verified
//
#include <hip/hip_runtime.h>
#include <math.h>

#define N_NODES 100000
#define N_EDGES 3200000
#define EDGE_DIM 8

typedef __attribute__((ext_vector_type(2))) float v2f;
typedef __attribute__((ext_vector_type(8))) float v8f;

// ---------- order-preserving float <-> uint for atomicMax segment-max ----------
__device__ __forceinline__ unsigned enc_f32(float f) {
    unsigned u = __float_as_uint(f);
    return (u & 0x80000000u) ? ~u : (u | 0x80000000u);
}
__device__ __forceinline__ float dec_f32(unsigned k) {
    return (k & 0x80000000u) ? __uint_as_float(k ^ 0x80000000u)
                             : __uint_as_float(~k);
}
#define ENC_NEG_INF 0x007FFFFFu   // enc_f32(-inf)

__device__ __forceinline__ v8f wmma_f32_k4(v2f a, v2f b, v8f c) {
#if __has_builtin(__builtin_amdgcn_wmma_f32_16x16x4_f32)
    return __builtin_amdgcn_wmma_f32_16x16x4_f32(false, a, false, b, (short)0, c,
                                                 false, false);
#else
    return c;   // host-pass placeholder only; device pass uses the builtin
#endif
}

// ---------------------------------------------------------------------------
// Fused projections: {Q,K,V,S}[n,HC] = X[n,CIN] @ {wq,wk,wv,ws} + bias
// Block = 128 threads = 4 waves; block covers 64 rows (4 x 16-row tiles).
// X tile staged coalesced into LDS. Weight panels staged into LDS as
// PRE-PAIRED float2: Wp[k/2][col] = {W[k][col], W[k+1][col]} (zero-padded to
// HC16 cols). A lane's B operand for K-step k0 (always even) is then ONE
// contiguous 8B-aligned ds_load_b64 -> lands in an even VGPR pair, no repack.
// A layout (ISA 7.12.2): lanes 0-15 M rows, v0={K0|K2}, v1={K1|K3} (hi lanes
// K2/K3). B symmetric with N across lanes. C/D: vgpr v = rows M=v / M=v+8.
// ---------------------------------------------------------------------------
template<int CIN, int HCC>
__global__ __launch_bounds__(128) void proj4_wmma(
        const float* __restrict__ X,
        const float* __restrict__ wq, const float* __restrict__ bq,
        const float* __restrict__ wk, const float* __restrict__ bk,
        const float* __restrict__ wv, const float* __restrict__ bv,
        const float* __restrict__ wsk, const float* __restrict__ bsk,
        float* __restrict__ Q, float* __restrict__ K,
        float* __restrict__ V, float* __restrict__ S, int nrows) {
    constexpr int HC16 = (HCC + 15) & ~15;
    constexpr int KP = CIN / 2;                 // paired K rows
    __shared__ v2f Wl[4][KP * HC16];
    __shared__ float Xl[64 * CIN];

    const int tid = threadIdx.x;
    const float* const wsrc[4] = {wq, wk, wv, wsk};
#pragma unroll
    for (int m = 0; m < 4; ++m) {
        for (int i = tid; i < KP * HC16; i += 128) {
            const int p = i / HC16, c = i - p * HC16;
            v2f w = {0.f, 0.f};
            if (c < HCC) {
                w.x = wsrc[m][(2 * p) * HCC + c];
                w.y = wsrc[m][(2 * p + 1) * HCC + c];
            }
            Wl[m][i] = w;
        }
    }
    const int row_base = blockIdx.x * 64;
    for (int i = tid; i < 64 * (CIN / 4); i += 128) {
        const int r = i / (CIN / 4);
        const int c4 = i - r * (CIN / 4);
        float4 v4 = make_float4(0.f, 0.f, 0.f, 0.f);
        if (row_base + r < nrows)
            v4 = ((const float4*)(X + (size_t)(row_base + r) * CIN))[c4];
        ((float4*)Xl)[i] = v4;
    }
    __syncthreads();

    const int wave = tid >> 5, lane = tid & 31;
    const int half = lane >> 4, lp = lane & 15;
    const int row0 = row_base + wave * 16;
    if (row0 >= nrows) return;              // wave-uniform: EXEC all-1 at WMMA

    const float* xa = Xl + (wave * 16 + lp) * CIN;

#pragma unroll 1
    for (int col0 = 0; col0 < HC16; col0 += 16) {
        const int col = col0 + lp;
        v8f aq = {}, ak = {}, av = {}, as = {};
#pragma unroll
        for (int kb = 0; kb < CIN; kb += 4) {
            const int k0 = kb + 2 * half;       // even
            const int p0 = (kb >> 1) + half;    // pair row index
            const v2f a  = *(const v2f*)(xa + k0);
            const v2f b0 = Wl[0][p0 * HC16 + col];
            const v2f b1 = Wl[1][p0 * HC16 + col];
            const v2f b2 = Wl[2][p0 * HC16 + col];
            const v2f b3 = Wl[3][p0 * HC16 + col];
            aq = wmma_f32_k4(a, b0, aq);
            ak = wmma_f32_k4(a, b1, ak);
            av = wmma_f32_k4(a, b2, av);
            as = wmma_f32_k4(a, b3, as);
        }
        if (col < HCC) {
            const float vq = bq[col], vk = bk[col], vv = bv[col], vs = bsk[col];
#pragma unroll
            for (int v = 0; v < 8; ++v) {
                const size_t r = (size_t)(row0 + v + 8 * half);
                Q[r * HCC + col] = aq[v] + vq;
                K[r * HCC + col] = ak[v] + vk;
                V[r * HCC + col] = av[v] + vv;
                S[r * HCC + col] = as[v] + vs;
            }
        }
    }
}

// ---------------------------------------------------------------------------
__global__ void seg_init(unsigned* __restrict__ mkey, float* __restrict__ den,
                         float* __restrict__ outp, long nmh, long nout) {
    const long stride = (long)gridDim.x * blockDim.x;
    const long i = (long)blockIdx.x * blockDim.x + threadIdx.x;
    for (long j = i; j < nmh; j += stride) { mkey[j] = ENC_NEG_INF; den[j] = 0.f; }
    for (long j = i; j < nout; j += stride) outp[j] = 0.f;
}

// ---------------------------------------------------------------------------
// Pass B: logits[e,h] = (q[dst] . (k[src] + edge_attr@we)) / sqrt(C)
//         + segment-max via ordered-uint atomicMax.
// ---------------------------------------------------------------------------
template<int H, int C>
__global__ __launch_bounds__(256) void edge_logits_t(
        const float* __restrict__ Q, const float* __restrict__ K,
        const float* __restrict__ ea, const int* __restrict__ ei,
        const float* __restrict__ we, float* __restrict__ elog,
        unsigned* __restrict__ mkey, float scale) {
    constexpr int HC = H * C;
    __shared__ float weS[EDGE_DIM * HC];
    for (int i = threadIdx.x; i < EDGE_DIM * HC; i += 256) weS[i] = we[i];
    __syncthreads();
    const long e = (long)blockIdx.x * blockDim.x + threadIdx.x;
    if (e >= N_EDGES) return;
    const int src = ei[e], dst = ei[N_EDGES + e];
    const float4 a0 = ((const float4*)ea)[e * 2];
    const float4 a1 = ((const float4*)ea)[e * 2 + 1];
    const float a[EDGE_DIM] = {a0.x, a0.y, a0.z, a0.w, a1.x, a1.y, a1.z, a1.w};
    const float* __restrict__ qrow = Q + (size_t)dst * HC;
    const float* __restrict__ krow = K + (size_t)src * HC;
#pragma unroll
    for (int h = 0; h < H; ++h) {
        float lg = 0.f;
#pragma unroll
        for (int c = 0; c < C; ++c) {
            const int j = h * C + c;
            float ev = 0.f;
#pragma unroll
            for (int d = 0; d < EDGE_DIM; ++d) ev += a[d] * weS[d * HC + j];
            lg += qrow[j] * (krow[j] + ev);
        }
        lg *= scale;
        elog[e * H + h] = lg;
        atomicMax(&mkey[(size_t)dst * H + h], enc_f32(lg));
    }
}

// ---------------------------------------------------------------------------
// Pass C: ex = exp(logit - max[dst]); store in-place; den[dst] += ex.
// ---------------------------------------------------------------------------
template<int H>
__global__ __launch_bounds__(256) void edge_exp_t(
        float* __restrict__ elog, const unsigned* __restrict__ mkey,
        float* __restrict__ den, const int* __restrict__ ei) {
    const long t = (long)blockIdx.x * blockDim.x + threadIdx.x;
    if (t >= (long)N_EDGES * H) return;
    const long e = t / H;
    const int  h = (int)(t - e * H);
    const int dst = ei[N_EDGES + e];
    const float ex = expf(elog[t] - dec_f32(mkey[(size_t)dst * H + h]));
    elog[t] = ex;
    atomicAdd(&den[(size_t)dst * H + h], ex);
}

// ---------------------------------------------------------------------------
// Pass D: out[dst] += ex * (v[src] + edge_attr@we)   (unnormalized)
// ---------------------------------------------------------------------------
template<int H, int C>
__global__ __launch_bounds__(256) void edge_aggr_t(
        const float* __restrict__ V, const float* __restrict__ elog,
        const float* __restrict__ ea, const int* __restrict__ ei,
        const float* __restrict__ we, float* __restrict__ outp) {
    constexpr int HC = H * C;
    __shared__ float weS[EDGE_DIM * HC];
    for (int i = threadIdx.x; i < EDGE_DIM * HC; i += 256) weS[i] = we[i];
    __syncthreads();
    const long e = (long)blockIdx.x * blockDim.x + threadIdx.x;
    if (e >= N_EDGES) return;
    const int src = ei[e], dst = ei[N_EDGES + e];
    const float4 a0 = ((const float4*)ea)[e * 2];
    const float4 a1 = ((const float4*)ea)[e * 2 + 1];
    const float a[EDGE_DIM] = {a0.x, a0.y, a0.z, a0.w, a1.x, a1.y, a1.z, a1.w};
    const float* __restrict__ vrow = V + (size_t)src * HC;
#pragma unroll
    for (int h = 0; h < H; ++h) {
        const float ex = elog[e * H + h];
#pragma unroll
        for (int c = 0; c < C; ++c) {
            const int j = h * C + c;
            float ev = 0.f;
#pragma unroll
            for (int d = 0; d < EDGE_DIM; ++d) ev += a[d] * weS[d * HC + j];
            atomicAdd(&outp[(size_t)dst * HC + j], ex * (vrow[j] + ev));
        }
    }
}

// ---------------------------------------------------------------------------
// Node gate: out /= den;  g = sigmoid([out, x_r, out - x_r] @ wbeta);
//            h = relu(g * x_r + (1 - g) * out)
// ---------------------------------------------------------------------------
template<int H, int C>
__global__ __launch_bounds__(256) void node_update_t(
        const float* __restrict__ Outp, const float* __restrict__ S,
        const float* __restrict__ den, const float* __restrict__ wbeta,
        float* __restrict__ hdst) {
    constexpr int HC = H * C;
    __shared__ float wbS[3 * HC];
    for (int i = threadIdx.x; i < 3 * HC; i += 256) wbS[i] = wbeta[i];
    __syncthreads();
    const int n = blockIdx.x * blockDim.x + threadIdx.x;
    if (n >= N_NODES) return;
    float acc = 0.f;
#pragma unroll
    for (int i = 0; i < HC; ++i) {
        const float dnm = den[(size_t)n * H + i / C] + 1e-16f;
        const float o = Outp[(size_t)n * HC + i] / dnm;
        const float xr = S[(size_t)n * HC + i];
        acc += o * wbS[i] + xr * wbS[HC + i] + (o - xr) * wbS[2 * HC + i];
    }
    const float g = 1.f / (1.f + expf(-acc));
#pragma unroll
    for (int i = 0; i < HC; ++i) {
        const float dnm = den[(size_t)n * H + i / C] + 1e-16f;
        const float o = Outp[(size_t)n * HC + i] / dnm;
        const float xr = S[(size_t)n * HC + i];
        const float v = g * xr + (1.f - g) * o;
        hdst[(size_t)n * HC + i] = fmaxf(v, 0.f);
    }
}

// ---------------------------------------------------------------------------
__global__ __launch_bounds__(256) void final_lin(
        const float* __restrict__ Hf, const float* __restrict__ wc,
        const float* __restrict__ bc, float* __restrict__ outp) {
    __shared__ float wcS[64];
    for (int i = threadIdx.x; i < 64; i += 256) wcS[i] = wc[i];
    __syncthreads();
    const int n = blockIdx.x * blockDim.x + threadIdx.x;
    if (n >= N_NODES) return;
    float acc = bc[0];
#pragma unroll
    for (int i = 0; i < 64; ++i) acc += Hf[(size_t)n * 64 + i] * wcS[i];
    outp[n] = acc;
}

// ---------------------------------------------------------------------------
template<int CIN, int H, int C>
static void run_layer(const float* X, void* const* d_in, int base,
                      float* Q, float* Kp, float* Vp, float* Sp, float* Outp,
                      unsigned* mkey, float* den, float* elog,
                      const int* ei, const float* ea, float* hdst,
                      hipStream_t stream) {
    constexpr int HC = H * C;
    const float* wq  = (const float*)d_in[base + 0];
    const float* bq  = (const float*)d_in[base + 1];
    const float* wk  = (const float*)d_in[base + 2];
    const float* bk  = (const float*)d_in[base + 3];
    const float* wv  = (const float*)d_in[base + 4];
    const float* bv  = (const float*)d_in[base + 5];
    const float* we  = (const float*)d_in[base + 6];
    const float* wsk = (const float*)d_in[base + 7];
    const float* bsk = (const float*)d_in[base + 8];
    const float* wbeta = (const float*)d_in[base + 9];

    const int pblocks = (N_NODES + 63) / 64;
    proj4_wmma<CIN, HC><<<pblocks, 128, 0, stream>>>(
        X, wq, bq, wk, bk, wv, bv, wsk, bsk, Q, Kp, Vp, Sp, N_NODES);

    seg_init<<<2048, 256, 0, stream>>>(mkey, den, Outp,
                                       (long)N_NODES * H, (long)N_NODES * HC);

    const float scale = 1.0f / sqrtf((float)C);
    const int eb = (N_EDGES + 255) / 256;
    edge_logits_t<H, C><<<eb, 256, 0, stream>>>(Q, Kp, ea, ei, we, elog, mkey, scale);

    const int eb2 = (int)(((long)N_EDGES * H + 255) / 256);
    edge_exp_t<H><<<eb2, 256, 0, stream>>>(elog, mkey, den, ei);

    edge_aggr_t<H, C><<<eb, 256, 0, stream>>>(Vp, elog, ea, ei, we, Outp);

    node_update_t<H, C><<<(N_NODES + 255) / 256, 256, 0, stream>>>(
        Outp, Sp, den, wbeta, hdst);
}

// ---------------------------------------------------------------------------
extern "C" void kernel_launch(void* const* d_in, const int* in_sizes, int n_in,
                              void* d_out, int out_size, void* d_ws, size_t ws_size,
                              hipStream_t stream) {
    (void)in_sizes; (void)n_in; (void)out_size; (void)ws_size;
    const float* x  = (const float*)d_in[0];
    const int*   ei = (const int*)d_in[1];
    const float* ea = (const float*)d_in[2];
    const float* wc = (const float*)d_in[33];
    const float* bc = (const float*)d_in[34];

    float* ws = (float*)d_ws;
    size_t off = 0;
    auto carve = [&](size_t nelem) -> float* {
        float* p = ws + off;
        off += (nelem + 63) & ~(size_t)63;
        return p;
    };
    float* Q    = carve((size_t)N_NODES * 64);
    float* Kp   = carve((size_t)N_NODES * 64);
    float* Vp   = carve((size_t)N_NODES * 64);
    float* Sp   = carve((size_t)N_NODES * 64);
    float* Outp = carve((size_t)N_NODES * 64);
    float* hA   = carve((size_t)N_NODES * 64);
    float* hB   = carve((size_t)N_NODES * 64);
    unsigned* mkey = (unsigned*)carve((size_t)N_NODES * 8);
    float* den  = carve((size_t)N_NODES * 8);
    float* elog = carve((size_t)N_EDGES * 8);

    run_layer<32, 8, 6>(x, d_in, 3,  Q, Kp, Vp, Sp, Outp, mkey, den, elog,
                        ei, ea, hA, stream);
    run_layer<48, 8, 3>(hA, d_in, 13, Q, Kp, Vp, Sp, Outp, mkey, den, elog,
                        ei, ea, hB, stream);
    run_layer<24, 1, 64>(hB, d_in, 23, Q, Kp, Vp, Sp, Outp, mkey, den, elog,
                         ei, ea, hA, stream);

    final_lin<<<(N_NODES + 255) / 256, 256, 0, stream>>>(hA, wc, bc, (float*)d_out);
}